// TripletLoss_4234837754126
// MI455X (gfx1250) — compile-verified
//
#include <hip/hip_runtime.h>

typedef __attribute__((ext_vector_type(2))) float v2f;
typedef __attribute__((ext_vector_type(8))) float v8f;

// Layout-agnostic wave32 sum using V_WMMA_F32_16X16X4_F32.
// A-fragment = {x, 0} per lane (64 values total), B = all-ones.
// C[m][n] = sum_k A[m][k]  =>  sum over column n (lanes n and n+16) = sum of
// ALL A values = sum over the 32 lanes of x. Broadcast with one shfl_xor(16).
__device__ __forceinline__ float wave_sum_wmma(float x) {
    v2f a; a[0] = x;    a[1] = 0.0f;
    v2f b; b[0] = 1.0f; b[1] = 1.0f;
    v8f c = {};
    c = __builtin_amdgcn_wmma_f32_16x16x4_f32(
            /*neg_a=*/false, a, /*neg_b=*/false, b,
            /*c_mod=*/(short)0, c, /*reuse_a=*/false, /*reuse_b=*/false);
    float s = ((c[0] + c[1]) + (c[2] + c[3])) + ((c[4] + c[5]) + (c[6] + c[7]));
    s += __shfl_xor(s, 16, 32);   // combine M=0..7 half with M=8..15 half
    return s;                     // every lane now holds the full wave sum
}

// One wave per row (D = 256). Lane L handles columns [4L,4L+4) and [128+4L,128+4L+4).
__global__ void __launch_bounds__(256)
triplet_partial(const float* __restrict__ A, const float* __restrict__ P,
                const float* __restrict__ Ng, const float* __restrict__ W,
                const float* __restrict__ marginp, float* __restrict__ partials,
                int N) {
    const int lane          = threadIdx.x & 31;
    const int waveInBlock   = threadIdx.x >> 5;
    const int wavesPerBlock = blockDim.x >> 5;
    const int globalWave    = blockIdx.x * wavesPerBlock + waveInBlock;
    const int numWaves      = gridDim.x * wavesPerBlock;

    const float margin = *marginp;

    const int c0 = lane * 4;
    const int c1 = 128 + lane * 4;

    // Per-lane exp(W) weights, hoisted out of the streaming loop.
    const float4 W0 = *(const float4*)(W + c0);
    const float4 W1 = *(const float4*)(W + c1);
    float4 w0, w1;
    w0.x = expf(W0.x); w0.y = expf(W0.y); w0.z = expf(W0.z); w0.w = expf(W0.w);
    w1.x = expf(W1.x); w1.y = expf(W1.y); w1.z = expf(W1.z); w1.w = expf(W1.w);

    float acc = 0.0f;

    for (int r = globalWave; r < N; r += numWaves) {
        const size_t base = (size_t)r * 256;
        const float4 a0 = *(const float4*)(A  + base + c0);
        const float4 a1 = *(const float4*)(A  + base + c1);
        const float4 p0 = *(const float4*)(P  + base + c0);
        const float4 p1 = *(const float4*)(P  + base + c1);
        const float4 n0 = *(const float4*)(Ng + base + c0);
        const float4 n1 = *(const float4*)(Ng + base + c1);

        float dp = 0.0f, dn = 0.0f, t;
        t = a0.x - p0.x; dp = __fmaf_rn(t * t, w0.x, dp);
        t = a0.y - p0.y; dp = __fmaf_rn(t * t, w0.y, dp);
        t = a0.z - p0.z; dp = __fmaf_rn(t * t, w0.z, dp);
        t = a0.w - p0.w; dp = __fmaf_rn(t * t, w0.w, dp);
        t = a1.x - p1.x; dp = __fmaf_rn(t * t, w1.x, dp);
        t = a1.y - p1.y; dp = __fmaf_rn(t * t, w1.y, dp);
        t = a1.z - p1.z; dp = __fmaf_rn(t * t, w1.z, dp);
        t = a1.w - p1.w; dp = __fmaf_rn(t * t, w1.w, dp);

        t = a0.x - n0.x; dn = __fmaf_rn(t * t, w0.x, dn);
        t = a0.y - n0.y; dn = __fmaf_rn(t * t, w0.y, dn);
        t = a0.z - n0.z; dn = __fmaf_rn(t * t, w0.z, dn);
        t = a0.w - n0.w; dn = __fmaf_rn(t * t, w0.w, dn);
        t = a1.x - n1.x; dn = __fmaf_rn(t * t, w1.x, dn);
        t = a1.y - n1.y; dn = __fmaf_rn(t * t, w1.y, dn);
        t = a1.z - n1.z; dn = __fmaf_rn(t * t, w1.z, dn);
        t = a1.w - n1.w; dn = __fmaf_rn(t * t, w1.w, dn);

        // Cross-lane reductions on the matrix pipe (EXEC is all-ones here:
        // the row loop is wave-uniform).
        const float d_pos = wave_sum_wmma(dp);
        const float d_neg = wave_sum_wmma(dn);

        const float loss = d_pos - d_neg + margin;
        acc += fmaxf(loss, 0.0f);   // identical value in every lane
    }

    __shared__ float lds[8];
    if (lane == 0) lds[waveInBlock] = acc;
    __syncthreads();
    if (threadIdx.x == 0) {
        float s = 0.0f;
        for (int i = 0; i < wavesPerBlock; ++i) s += lds[i];   // fixed order
        partials[blockIdx.x] = s;
    }
}

__global__ void __launch_bounds__(256)
reduce_final(const float* __restrict__ partials, int B, float invN,
             float* __restrict__ out) {
    __shared__ double lds[256];
    double s = 0.0;
    for (int i = threadIdx.x; i < B; i += blockDim.x) s += (double)partials[i];
    lds[threadIdx.x] = s;
    __syncthreads();
    for (int off = 128; off > 0; off >>= 1) {
        if (threadIdx.x < (unsigned)off) lds[threadIdx.x] += lds[threadIdx.x + off];
        __syncthreads();
    }
    if (threadIdx.x == 0) out[0] = (float)(lds[0] * (double)invN);
}

extern "C" void kernel_launch(void* const* d_in, const int* in_sizes, int n_in,
                              void* d_out, int out_size, void* d_ws, size_t ws_size,
                              hipStream_t stream) {
    const float* A  = (const float*)d_in[0];   // anchor   [N,256]
    const float* P  = (const float*)d_in[1];   // positive [N,256]
    const float* Ng = (const float*)d_in[2];   // negative [N,256]
    const float* W  = (const float*)d_in[3];   // W        [256]
    const float* M  = (const float*)d_in[4];   // margin   scalar

    const int D = in_sizes[3];                 // 256
    const int N = in_sizes[0] / D;             // 262144

    float* partials = (float*)d_ws;
    const int BLOCKS = 1024;                   // 8192 waves -> 32 rows/wave

    triplet_partial<<<BLOCKS, 256, 0, stream>>>(A, P, Ng, W, M, partials, N);
    reduce_final<<<1, 256, 0, stream>>>(partials, BLOCKS, 1.0f / (float)N,
                                        (float*)d_out);
}